// LengthRegulator_12360915877981
// MI455X (gfx1250) — compile-verified
//
#include <hip/hip_runtime.h>

// LengthRegulator for MI455X (gfx1250).
// Pure gather/copy -> bandwidth-bound. Data path: async global->LDS->global
// DMA (B128 per lane, ASYNCcnt-tracked) per CDNA5 ISA ch.10 / CDNA5 ISA.
// Output stores are non-temporal (TH_STORE_NT) so the ~226MB write stream does
// not evict the 64MB encoder working set from the 192MB L2.

#define BATCH 32
#define DIMV4 128          // DIM=512 floats = 128 float4
#define WAVES 8            // waves (frames) per gather block
#define GATHER_TPB (32 * WAVES)

// ---------------------------------------------------------------------------
// Kernel 1: per-batch inclusive scan of durations -> cum[b][n] (int32)
// One 256-thread block per batch row.
// ---------------------------------------------------------------------------
__global__ __launch_bounds__(256)
void lr_scan(const int* __restrict__ dur, int* __restrict__ cum, int N) {
    const int b   = blockIdx.x;
    const int tid = threadIdx.x;
    const int items = (N + 255) >> 8;          // per-thread elements (4 for N=1024)
    const int base  = tid * items;

    int vals[16];                              // supports N up to 4096
    int s = 0;
    for (int i = 0; i < items; ++i) {
        int g = base + i;
        int v = (g < N) ? dur[(size_t)b * N + g] : 0;
        s += v;
        vals[i] = s;                           // inclusive within thread
    }

    __shared__ int ts[256];
    ts[tid] = s;
    __syncthreads();
    // Hillis-Steele inclusive scan over the 256 thread sums.
    for (int off = 1; off < 256; off <<= 1) {
        int v = (tid >= off) ? ts[tid - off] : 0;
        __syncthreads();
        ts[tid] += v;
        __syncthreads();
    }
    const int excl = (tid == 0) ? 0 : ts[tid - 1];

    for (int i = 0; i < items; ++i) {
        int g = base + i;
        if (g < N) cum[(size_t)b * N + g] = excl + vals[i];
    }
}

// ---------------------------------------------------------------------------
// Kernel 2: per-(b,t) source index via binary search (searchsorted right),
// plus the tuple's second output: output_mask as 0.0/1.0 float.
// idx = -1 marks frames at/past the sequence end (to be zero-filled).
// ---------------------------------------------------------------------------
__global__ __launch_bounds__(256)
void lr_idx(const int* __restrict__ cum, int* __restrict__ idx,
            float* __restrict__ mask_out, int N, int T, int write_mask) {
    const long long i = (long long)blockIdx.x * blockDim.x + threadIdx.x;
    if (i >= (long long)BATCH * T) return;
    const int b = (int)(i / T);
    const int t = (int)(i % T);

    const int* c = cum + (size_t)b * N;
    const int total = c[N - 1];

    // first j with c[j] > t  ==  #{c <= t}  (searchsorted side='right')
    int lo = 0, hi = N;
    while (lo < hi) {
        int mid = (lo + hi) >> 1;
        if (c[mid] <= t) lo = mid + 1; else hi = mid;
    }
    int id = (lo < N - 1) ? lo : (N - 1);

    const bool valid = (t < total);
    idx[i] = valid ? id : -1;
    if (write_mask) mask_out[i] = valid ? 1.0f : 0.0f;
}

// ---------------------------------------------------------------------------
// Kernel 3: row gather via async global->LDS->global DMA (CDNA5 B128 async).
// grid = (ceil(T/WAVES), BATCH), block = 256 threads = 8 waves.
// Each wave copies one 2KB row: 4x global_load_async_to_lds_b128 (16B/lane),
// s_wait_asynccnt 0, then 4x global_store_async_from_lds_b128 (non-temporal).
// t is wave-uniform, so EXEC is all-ones inside the async region.
// ---------------------------------------------------------------------------
__global__ __launch_bounds__(GATHER_TPB)
void lr_gather(const float* __restrict__ enc, const int* __restrict__ idx,
               float* __restrict__ out, int N, int T) {
    __shared__ float4 stage[WAVES][DIMV4];     // 2KB per wave, 16KB per block

    const int b    = blockIdx.y;
    const int wave = threadIdx.x >> 5;
    const int lane = threadIdx.x & 31;
    const int t    = blockIdx.x * WAVES + wave;
    if (t >= T) return;

    const int id = idx[(size_t)b * T + t];
    float4* dst = (float4*)out + ((size_t)b * T + t) * DIMV4;

    if (id >= 0) {
        const float4* src = (const float4*)enc + ((size_t)b * N + id) * DIMV4;
        // LDS byte offset of this lane's float4 slot (generic addr low 32 bits
        // == LDS offset per the aperture mapping).
        const unsigned lbase =
            (unsigned)(uintptr_t)(void*)&stage[wave][lane];

        #pragma unroll
        for (int k = 0; k < 4; ++k) {
            unsigned laddr = lbase + (unsigned)(k * 32 * sizeof(float4));
            unsigned long long gaddr =
                (unsigned long long)(uintptr_t)(const void*)(src + k * 32 + lane);
            asm volatile("global_load_async_to_lds_b128 %0, %1, off"
                         :: "v"(laddr), "v"(gaddr) : "memory");
        }
        asm volatile("s_wait_asynccnt 0" ::: "memory");
        #pragma unroll
        for (int k = 0; k < 4; ++k) {
            unsigned laddr = lbase + (unsigned)(k * 32 * sizeof(float4));
            unsigned long long gaddr =
                (unsigned long long)(uintptr_t)(void*)(dst + k * 32 + lane);
            asm volatile("global_store_async_from_lds_b128 %0, %1, off th:TH_STORE_NT"
                         :: "v"(gaddr), "v"(laddr) : "memory");
        }
        asm volatile("s_wait_asynccnt 0" ::: "memory");
    } else {
        // Invalid frame: explicit zero-fill (d_out is poisoned, not zeroed).
        // Non-temporal: write-once stream.
        #pragma unroll
        for (int k = 0; k < 4; ++k) {
            float* p = (float*)(dst + k * 32 + lane);
            __builtin_nontemporal_store(0.0f, p + 0);
            __builtin_nontemporal_store(0.0f, p + 1);
            __builtin_nontemporal_store(0.0f, p + 2);
            __builtin_nontemporal_store(0.0f, p + 3);
        }
    }
}

// ---------------------------------------------------------------------------
// Launch
// ---------------------------------------------------------------------------
extern "C" void kernel_launch(void* const* d_in, const int* in_sizes, int n_in,
                              void* d_out, int out_size, void* d_ws, size_t ws_size,
                              hipStream_t stream) {
    const float* enc = (const float*)d_in[0];   // (B, N, DIM) f32
    const int*   dur = (const int*)d_in[1];     // (B, N) int
    // d_in[2] (output_mask) not needed: it equals (t < cum[b][N-1]).

    const int BN  = in_sizes[1];                // B*N
    const int N   = BN / BATCH;
    const int BT  = in_sizes[2];                // B*T
    const int T   = BT / BATCH;
    const int DIM = in_sizes[0] / BN;           // 512

    float* out  = (float*)d_out;
    int*   cum  = (int*)d_ws;                   // B*N ints
    int*   idx  = cum + BN;                     // B*T ints

    // Tuple output layout: [expanded (B*T*DIM) | output_mask (B*T)] if room.
    const long long need = (long long)BT * DIM + BT;
    const int write_mask = (out_size >= need) ? 1 : 0;
    float* mask_out = out + (size_t)BT * DIM;

    if (T <= 0) return;

    lr_scan<<<BATCH, 256, 0, stream>>>(dur, cum, N);
    lr_idx<<<(BT + 255) / 256, 256, 0, stream>>>(cum, idx, mask_out, N, T, write_mask);

    dim3 grid((T + WAVES - 1) / WAVES, BATCH);
    lr_gather<<<grid, GATHER_TPB, 0, stream>>>(enc, idx, out, N, T);
    (void)n_in; (void)ws_size; (void)DIM;
}